// GNNModel_26268019983050
// MI455X (gfx1250) — compile-verified
//
#include <hip/hip_runtime.h>
#include <math.h>

#define N_NODES 100000

typedef float v2f __attribute__((ext_vector_type(2)));
typedef float v8f __attribute__((ext_vector_type(8)));

// Non-returning fp32 atomic add -> GLOBAL_ATOMIC_ADD_F32 (no CAS loop).
__device__ __forceinline__ void atomicAddF32(float* p, float v) {
    __hip_atomic_fetch_add(p, v, __ATOMIC_RELAXED, __HIP_MEMORY_SCOPE_AGENT);
}

__global__ void zero4_kernel(float4* __restrict__ p, int n4) {
    int i = blockIdx.x * blockDim.x + threadIdx.x;
    if (i < n4) p[i] = make_float4(0.f, 0.f, 0.f, 0.f);
}

// Edge-parallel degree accumulation (float counts, like the reference).
__global__ void degree_kernel(const int* __restrict__ src, const int* __restrict__ dst,
                              float* __restrict__ dout, float* __restrict__ din, int E) {
    int e = blockIdx.x * blockDim.x + threadIdx.x;
    if (e < E) {
        atomicAddF32(&dout[src[e]], 1.0f);
        atomicAddF32(&din[dst[e]], 1.0f);
    }
}

__global__ void rsqrt_kernel(float* __restrict__ d, int n) {
    int i = blockIdx.x * blockDim.x + threadIdx.x;
    if (i < n) d[i] = rsqrtf(fmaxf(d[i], 1.0f));
}

// One thread per (edge, float4 chunk of the 64-wide feature row).
// Fuses the source-side D_out^{-1/2} scale into the gather:
//   agg[dst] += x[src] * doutI[src]
__global__ void scatter_scaled_kernel(const float* __restrict__ x,
                                      const float* __restrict__ doutI,
                                      const int* __restrict__ src,
                                      const int* __restrict__ dst,
                                      float* __restrict__ agg, int E) {
    int idx = blockIdx.x * blockDim.x + threadIdx.x;
    if (idx >= E * 16) return;
    int e = idx >> 4;
    int c = (idx & 15) << 2;
    int s = src[e];
    const float  sc = doutI[s];
    const float4 v  = *(const float4*)(x + (long)s * 64 + c);
    float* o = agg + (long)dst[e] * 64 + c;
    atomicAddF32(o + 0, v.x * sc);
    atomicAddF32(o + 1, v.y * sc);
    atomicAddF32(o + 2, v.z * sc);
    atomicAddF32(o + 3, v.w * sc);
}

// out[m, n] = act( (A[m,:] * dinI[m]) @ W[:, n] + bias[n] )
// A: [N,64], W: [64, NCOL] row-major, out: [N, NCOL].
// One wave32 per 16x16 tile, K=64 swept in 16 V_WMMA_F32_16X16X4_F32 steps.
// NCOL compile-time so all A/B loads use immediate offsets from fixed bases.
// ACT: 0 = identity, 1 = tanh, 2 = relu
template <int NCOL, int ACT>
__global__ void wmma_gemm_kernel(const float* __restrict__ A, const float* __restrict__ dinI,
                                 const float* __restrict__ W, const float* __restrict__ bias,
                                 float* __restrict__ out) {
    const int wave    = blockIdx.x * (blockDim.x >> 5) + (threadIdx.x >> 5);
    const int tiles_x = NCOL >> 4;
    const int m0      = (wave / tiles_x) << 4;
    const int n0      = (wave % tiles_x) << 4;
    const int lane    = threadIdx.x & 31;
    const int half    = lane >> 4;   // 0: this lane carries K=0,1 ; 1: K=2,3 per step
    const int l16     = lane & 15;

    const int   arow  = m0 + l16;
    const float scale = dinI[arow];
    const float* ap = A + (long)arow * 64 + 2 * half;          // A[arow][k0], A[arow][k0+1]
    const float* bp = W + (long)(2 * half) * NCOL + n0 + l16;  // W[k0][n0+l16]

    v8f acc = {};
#pragma unroll
    for (int k = 0; k < 16; ++k) {
        v2f a;
        a.x = ap[k * 4 + 0] * scale;
        a.y = ap[k * 4 + 1] * scale;
        v2f b;
        b.x = bp[(k * 4 + 0) * NCOL];
        b.y = bp[(k * 4 + 1) * NCOL];
        acc = __builtin_amdgcn_wmma_f32_16x16x4_f32(
            /*neg_a=*/false, a, /*neg_b=*/false, b,
            /*c_mod=*/(short)0, acc, /*reuse_a=*/false, /*reuse_b=*/false);
    }

    const int   ocol = n0 + l16;
    const float bval = bias[ocol];
#pragma unroll
    for (int r = 0; r < 8; ++r) {
        int   orow = m0 + r + 8 * half;   // C/D layout: VGPR r -> M=r (lanes 0-15), M=r+8 (lanes 16-31)
        float v    = acc[r] + bval;
        if (ACT == 1) v = tanhf(v);
        if (ACT == 2) v = fmaxf(v, 0.0f);
        out[(long)orow * NCOL + ocol] = v;
    }
}

// z_adj = m + exp(m)*eps ; z_log_std = m   (z_mean already stored by the GEMM)
__global__ void reparam_kernel(const float* __restrict__ zm, const float* __restrict__ eps,
                               float* __restrict__ zadj, float* __restrict__ zlog, int n) {
    int i = blockIdx.x * blockDim.x + threadIdx.x;
    if (i < n) {
        float m = zm[i];
        zadj[i] = m + expf(m) * eps[i];
        zlog[i] = m;
    }
}

static inline int cdiv(int a, int b) { return (a + b - 1) / b; }

extern "C" void kernel_launch(void* const* d_in, const int* in_sizes, int n_in,
                              void* d_out, int out_size, void* d_ws, size_t ws_size,
                              hipStream_t stream) {
    const float* feat = (const float*)d_in[0];
    const int*   src  = (const int*)d_in[1];
    const int*   dst  = (const int*)d_in[2];
    const float* eps  = (const float*)d_in[3];
    const float* W1   = (const float*)d_in[4];
    const float* b1   = (const float*)d_in[5];
    const float* W2   = (const float*)d_in[6];
    const float* b2   = (const float*)d_in[7];
    const float* Wm   = (const float*)d_in[8];
    const float* bm   = (const float*)d_in[9];

    const int E = in_sizes[1];
    const int N = N_NODES;

    float* A     = (float*)d_ws;            // [N*64] aggregation buffer
    float* H     = A + (size_t)N * 64;      // [N*64] activations
    float* deg   = H + (size_t)N * 64;      // [2*N]: d_out | d_in
    float* doutI = deg;
    float* dinI  = deg + N;

    float* zadj = (float*)d_out;                     // [N*32]
    float* zlog = zadj + (size_t)N * 32;             // [N*32]
    float* zm   = zadj + (size_t)2 * N * 32;         // [N*32]

    const int nf = N * 64;

    // Degrees and normalizers
    zero4_kernel<<<cdiv(2 * N / 4, 256), 256, 0, stream>>>((float4*)deg, 2 * N / 4);
    degree_kernel<<<cdiv(E, 256), 256, 0, stream>>>(src, dst, doutI, dinI, E);
    rsqrt_kernel<<<cdiv(2 * N, 256), 256, 0, stream>>>(deg, 2 * N);

    // Layer 1: H = tanh( conv(feat) @ W1 + b1 )
    zero4_kernel<<<cdiv(nf / 4, 256), 256, 0, stream>>>((float4*)A, nf / 4);
    scatter_scaled_kernel<<<cdiv(E * 16, 256), 256, 0, stream>>>(feat, doutI, src, dst, A, E);
    wmma_gemm_kernel<64, 1><<<(N / 16) * (64 / 16) / 4, 128, 0, stream>>>(A, dinI, W1, b1, H);

    // Layer 2: H = relu( conv(H) @ W2 + b2 )
    zero4_kernel<<<cdiv(nf / 4, 256), 256, 0, stream>>>((float4*)A, nf / 4);
    scatter_scaled_kernel<<<cdiv(E * 16, 256), 256, 0, stream>>>(H, doutI, src, dst, A, E);
    wmma_gemm_kernel<64, 2><<<(N / 16) * (64 / 16) / 4, 128, 0, stream>>>(A, dinI, W2, b2, H);

    // Layer 3: zm = conv(H) @ Wm + bm   (z_mean == z_log_std, computed once)
    zero4_kernel<<<cdiv(nf / 4, 256), 256, 0, stream>>>((float4*)A, nf / 4);
    scatter_scaled_kernel<<<cdiv(E * 16, 256), 256, 0, stream>>>(H, doutI, src, dst, A, E);
    wmma_gemm_kernel<32, 0><<<(N / 16) * (32 / 16) / 4, 128, 0, stream>>>(A, dinI, Wm, bm, zm);

    // Reparameterization tail
    reparam_kernel<<<cdiv(N * 32, 256), 256, 0, stream>>>(zm, eps, zadj, zlog, N * 32);
}